// DecoderLayer_70111046140631
// MI455X (gfx1250) — compile-verified
//
#include <hip/hip_runtime.h>
#include <math.h>

// Problem constants (from reference)
#define B_  2
#define S_  2048
#define D_  1024
#define H_  16
#define DK_ 64
#define FF_ 4096
#define BS_ (B_ * S_)   // 4096 rows when batch is folded into M

typedef float v2f __attribute__((ext_vector_type(2)));
typedef float v4f __attribute__((ext_vector_type(4)));
typedef float v8f __attribute__((ext_vector_type(8)));

// fp32 WMMA:  D(16x16,f32) = A(16x4,f32) * B(4x16,f32) + C
// A frag (v2f): lane holds row = lane&15; a.x = K(2*(lane>>4)), a.y = K(2*(lane>>4)+1)
// B frag (v2f): lane holds col = lane&15; b.x = K(2*(lane>>4)), b.y = K(2*(lane>>4)+1)
// C/D (v8f):    element r -> row = r + 8*(lane>>4), col = lane&15
static __device__ __forceinline__ v8f wmma_f32(v2f a, v2f b, v8f c) {
  return __builtin_amdgcn_wmma_f32_16x16x4_f32(false, a, false, b, (short)0, c,
                                               false, false);
}

// CDNA5 async copy: global -> LDS, 16B per lane, tracked by ASYNCcnt.
// VDST = LDS byte offset (generic LDS pointer truncates to offset per ISA),
// VADDR = 64-bit global address.
static __device__ __forceinline__ void async_ld128(unsigned lds,
                                                   const float* g) {
  asm volatile("global_load_async_to_lds_b128 %0, %1, off"
               :: "v"(lds), "v"(g) : "memory");
}
static __device__ __forceinline__ void async_wait0() {
  asm volatile("s_wait_asynccnt 0x0" ::: "memory");
}

// ---------------------------------------------------------------------------
// GEMM:  C[M,N] = A[M,K] @ W[K,N] + bias[N]   (optional ReLU)
// Block = 256 threads = 8 waves; block tile = 128(M) x 64(N); wave tile 16x64.
// W k-chunks (32x64) stream through a DOUBLE-BUFFERED LDS tile via
// global_load_async_to_lds_b128: fetch of chunk t+1 overlaps WMMAs on chunk t.
// ---------------------------------------------------------------------------
#define KC  32
#define LDP 72   // padded LDS row stride (floats): 16B-aligned, conflict-light

__global__ __launch_bounds__(256) void gemm_wmma(
    const float* __restrict__ A, const float* __restrict__ W,
    const float* __restrict__ bias, float* __restrict__ C,
    int M, int N, int K, int relu) {
  __shared__ float wt[2][KC][LDP];

  const int tid  = threadIdx.x;
  const int wave = tid >> 5;
  const int lane = tid & 31;
  const int half = lane >> 4;
  const int l16  = lane & 15;
  const int row0 = blockIdx.y * 128 + wave * 16;
  const int col0 = blockIdx.x * 64;

  // per-thread async-copy slots: two 16B pieces of the 32x64 chunk
  const int p0 = tid * 4;
  const int r0 = p0 >> 6, c0 = p0 & 63;
  const int p1 = p0 + 1024;
  const int r1 = p1 >> 6, c1 = p1 & 63;
  const unsigned ldsA[2] = {(unsigned)(size_t)&wt[0][r0][c0],
                            (unsigned)(size_t)&wt[1][r0][c0]};
  const unsigned ldsB[2] = {(unsigned)(size_t)&wt[0][r1][c1],
                            (unsigned)(size_t)&wt[1][r1][c1]};

  v8f acc[4] = {};

  const int nchunk = K / KC;
  // prologue: fetch chunk 0 into buffer 0
  async_ld128(ldsA[0], W + (size_t)r0 * N + col0 + c0);
  async_ld128(ldsB[0], W + (size_t)r1 * N + col0 + c1);
  async_wait0();
  __syncthreads();

  for (int ch = 0; ch < nchunk; ++ch) {
    const int k0  = ch * KC;
    const int cur = ch & 1;

    // kick off next chunk into the other buffer (overlaps with compute)
    if (ch + 1 < nchunk) {
      const int kn = k0 + KC;
      async_ld128(ldsA[cur ^ 1], W + (size_t)(kn + r0) * N + col0 + c0);
      async_ld128(ldsB[cur ^ 1], W + (size_t)(kn + r1) * N + col0 + c1);
    }

    // batch-preload the 8 A fragments of this chunk (pipelined b64 loads)
    v2f a[8];
    const float* ap = A + (size_t)(row0 + l16) * K + k0 + half * 2;
#pragma unroll
    for (int j = 0; j < 8; ++j) a[j] = *(const v2f*)(ap + j * 4);

#pragma unroll
    for (int kc = 0; kc < KC; kc += 4) {
      const float* wp = &wt[cur][kc + half * 2][l16];
      v2f bf[4];
#pragma unroll
      for (int n = 0; n < 4; ++n) {
        bf[n].x = wp[n * 16];
        bf[n].y = wp[LDP + n * 16];
      }
#pragma unroll
      for (int n = 0; n < 4; ++n) acc[n] = wmma_f32(a[kc >> 2], bf[n], acc[n]);
    }

    if (ch + 1 < nchunk) {
      async_wait0();      // next chunk's LDS writes complete (this wave)
      __syncthreads();    // ... and visible to / ordered with all waves
    }
  }

#pragma unroll
  for (int n = 0; n < 4; ++n) {
    const int col = col0 + n * 16 + l16;
    const float bv = bias[col];
#pragma unroll
    for (int r = 0; r < 8; ++r) {
      const int row = row0 + r + 8 * half;
      float v = acc[n][r] + bv;
      if (relu) v = fmaxf(v, 0.0f);
      C[(size_t)row * N + col] = v;
    }
  }
}

// ---------------------------------------------------------------------------
// Flash-style attention, one wave per (b, h, 16-query tile).
// Q,K,V,O are [B*S, D] with head h occupying columns h*64 .. h*64+63.
// Online softmax; reference quirk: mask added before the 1/sqrt(dk) scale
// (identical with -inf masking). Fully masked key tiles are skipped.
// ---------------------------------------------------------------------------
__global__ __launch_bounds__(32) void attn_wmma(
    const float* __restrict__ Q, const float* __restrict__ Km,
    const float* __restrict__ V, float* __restrict__ O, int causal) {
  __shared__ float tile[16][18];   // stride 18: keeps v2f reads 8B-aligned
  __shared__ float mrow[16], lrow[16], frow[16];

  const int lane = threadIdx.x;
  const int half = lane >> 4;
  const int l16  = lane & 15;
  const int qt = blockIdx.x;
  const int h  = blockIdx.y;
  const int b  = blockIdx.z;
  const int q0 = qt * 16;
  const int hoff = h * DK_;
  const size_t qbase = ((size_t)b * S_ + q0) * D_;

  // preload all 16 Q fragments (dk = 64 -> 16 chunks of 4)
  v2f qf[16];
#pragma unroll
  for (int c = 0; c < 16; ++c)
    qf[c] = *(const v2f*)(Q + qbase + (size_t)l16 * D_ + hoff + c * 4 + half * 2);

  if (lane < 16) { mrow[lane] = -__builtin_inff(); lrow[lane] = 0.0f; }
  __syncthreads();

  v8f acc[4] = {};

  const int nkt = causal ? (qt + 1) : (S_ / 16);
  for (int kt = 0; kt < nkt; ++kt) {
    const int key0 = kt * 16;
    const size_t kbase = ((size_t)b * S_ + key0) * D_ + hoff;

    // scores S = Q Kt  (16 fp32 WMMAs)
    v8f s = {};
#pragma unroll
    for (int c = 0; c < 16; ++c) {
      v2f bk = *(const v2f*)(Km + kbase + (size_t)l16 * D_ + c * 4 + half * 2);
      s = wmma_f32(qf[c], bk, s);
    }

    // mask + scale, spill C-layout tile to LDS
#pragma unroll
    for (int r = 0; r < 8; ++r) {
      const int qrow = q0 + r + 8 * half;
      const int key  = key0 + l16;
      float v = s[r];
      if (causal && key > qrow) v = -__builtin_inff();
      tile[r + 8 * half][l16] = v * 0.125f;   // 1/sqrt(64)
    }
    __syncthreads();

    // online softmax: lanes 0..15 each own one query row
    if (lane < 16) {
      float mold = mrow[lane];
      float rmax = -__builtin_inff();
#pragma unroll
      for (int j = 0; j < 16; ++j) rmax = fmaxf(rmax, tile[lane][j]);
      const float mnew = fmaxf(mold, rmax);
      const float f = __expf(mold - mnew);
      float sum = 0.0f;
#pragma unroll
      for (int j = 0; j < 16; ++j) {
        const float e = __expf(tile[lane][j] - mnew);
        tile[lane][j] = e;
        sum += e;
      }
      lrow[lane] = lrow[lane] * f + sum;
      mrow[lane] = mnew;
      frow[lane] = f;
    }
    __syncthreads();

    // rescale running accumulators by exp(m_old - m_new)
    float fr[8];
#pragma unroll
    for (int r = 0; r < 8; ++r) fr[r] = frow[r + 8 * half];
#pragma unroll
    for (int n = 0; n < 4; ++n)
#pragma unroll
      for (int r = 0; r < 8; ++r) acc[n][r] *= fr[r];

    // acc += P @ V  (16 fp32 WMMAs); P re-read from LDS in A layout
#pragma unroll
    for (int c = 0; c < 4; ++c) {
      v2f pa = *(const v2f*)&tile[l16][c * 4 + half * 2];
      const float* vp = V + kbase + (size_t)(c * 4 + half * 2) * D_;
#pragma unroll
      for (int n = 0; n < 4; ++n) {
        v2f bv;
        bv.x = vp[n * 16 + l16];
        bv.y = vp[D_ + n * 16 + l16];
        acc[n] = wmma_f32(pa, bv, acc[n]);
      }
    }
    __syncthreads();
  }

  // finalize: divide by softmax denominator, write [B*S, D]
  float lr[8];
#pragma unroll
  for (int r = 0; r < 8; ++r) lr[r] = 1.0f / lrow[r + 8 * half];
#pragma unroll
  for (int n = 0; n < 4; ++n)
#pragma unroll
    for (int r = 0; r < 8; ++r)
      O[qbase + (size_t)(r + 8 * half) * D_ + hoff + n * 16 + l16] =
          acc[n][r] * lr[r];
}

// ---------------------------------------------------------------------------
// Sequence-dim norm (reference quirk: divide by UNBIASED VARIANCE, not std),
// fused with the residual add:  out = ((x+res) - mean_s) / var_s
// One block per 64 consecutive feature columns of one batch element.
// ---------------------------------------------------------------------------
__global__ __launch_bounds__(256) void seqnorm(
    const float* __restrict__ x, const float* __restrict__ res,
    float* __restrict__ out) {
  __shared__ float psum[4][64], psq[4][64];
  __shared__ float mean_s[64], inv_s[64];

  const int tx = threadIdx.x & 63;
  const int ty = threadIdx.x >> 6;
  const int d  = blockIdx.x * 64 + tx;
  const size_t base = (size_t)blockIdx.y * S_ * D_ + d;

  float sum = 0.0f, sq = 0.0f;
  for (int s = ty; s < S_; s += 4) {
    const float v = x[base + (size_t)s * D_] + res[base + (size_t)s * D_];
    sum += v; sq += v * v;
  }
  psum[ty][tx] = sum; psq[ty][tx] = sq;
  __syncthreads();
  if (ty == 0) {
    const float ts = psum[0][tx] + psum[1][tx] + psum[2][tx] + psum[3][tx];
    const float tq = psq[0][tx] + psq[1][tx] + psq[2][tx] + psq[3][tx];
    const float m  = ts / (float)S_;
    const float var = (tq - ts * m) / (float)(S_ - 1);   // unbiased
    mean_s[tx] = m;
    inv_s[tx]  = 1.0f / var;                              // divide by VARIANCE
  }
  __syncthreads();
  const float m  = mean_s[tx];
  const float iv = inv_s[tx];
  for (int s = ty; s < S_; s += 4) {
    const float v = x[base + (size_t)s * D_] + res[base + (size_t)s * D_];
    out[base + (size_t)s * D_] = (v - m) * iv;
  }
}

// ---------------------------------------------------------------------------
// Host-side orchestration
// ---------------------------------------------------------------------------
extern "C" void kernel_launch(void* const* d_in, const int* in_sizes, int n_in,
                              void* d_out, int out_size, void* d_ws,
                              size_t ws_size, hipStream_t stream) {
  (void)in_sizes; (void)n_in; (void)out_size; (void)ws_size;
  const float* dec = (const float*)d_in[0];
  const float* enc = (const float*)d_in[1];
  const float* Wq1 = (const float*)d_in[2];
  const float* Wk1 = (const float*)d_in[3];
  const float* Wv1 = (const float*)d_in[4];
  const float* bq1 = (const float*)d_in[5];
  const float* bk1 = (const float*)d_in[6];
  const float* bv1 = (const float*)d_in[7];
  const float* Wq2 = (const float*)d_in[8];
  const float* Wk2 = (const float*)d_in[9];
  const float* Wv2 = (const float*)d_in[10];
  const float* bq2 = (const float*)d_in[11];
  const float* bk2 = (const float*)d_in[12];
  const float* bv2 = (const float*)d_in[13];
  const float* W1  = (const float*)d_in[14];
  const float* b1  = (const float*)d_in[15];
  const float* W2  = (const float*)d_in[16];
  const float* b2  = (const float*)d_in[17];
  float* out = (float*)d_out;
  float* ws  = (float*)d_ws;

  const size_t SZ = (size_t)BS_ * D_;          // 16 MB per activation buffer
  float* q  = ws;                              // [4096,1024]
  float* k  = ws + SZ;
  float* v  = ws + 2 * SZ;
  float* t  = ws + 3 * SZ;                     // attention output
  float* x1 = ws + 4 * SZ;                     // sublayer-1 output
  float* hb = ws + 5 * SZ;                     // FFN hidden [4096,4096], 64 MB

  const dim3 gp(D_ / 64, BS_ / 128);           // projection GEMMs
  const dim3 gf(FF_ / 64, BS_ / 128);          // FFN1 GEMM
  const dim3 ga(S_ / 16, H_, B_);              // attention waves
  const dim3 gn(D_ / 64, B_);                  // seq-norm

  // sublayer 1: causal self-attention
  gemm_wmma<<<gp, 256, 0, stream>>>(dec, Wq1, bq1, q, BS_, D_, D_, 0);
  gemm_wmma<<<gp, 256, 0, stream>>>(dec, Wk1, bk1, k, BS_, D_, D_, 0);
  gemm_wmma<<<gp, 256, 0, stream>>>(dec, Wv1, bv1, v, BS_, D_, D_, 0);
  attn_wmma<<<ga, 32, 0, stream>>>(q, k, v, t, 1);
  seqnorm<<<gn, 256, 0, stream>>>(t, dec, x1);

  // sublayer 2: cross-attention (q from x1, k/v from encoder)
  gemm_wmma<<<gp, 256, 0, stream>>>(x1, Wq2, bq2, q, BS_, D_, D_, 0);
  gemm_wmma<<<gp, 256, 0, stream>>>(enc, Wk2, bk2, k, BS_, D_, D_, 0);
  gemm_wmma<<<gp, 256, 0, stream>>>(enc, Wv2, bv2, v, BS_, D_, D_, 0);
  attn_wmma<<<ga, 32, 0, stream>>>(q, k, v, t, 0);
  float* x2 = q;                               // q no longer needed
  seqnorm<<<gn, 256, 0, stream>>>(t, x1, x2);

  // sublayer 3: FFN
  gemm_wmma<<<gf, 256, 0, stream>>>(x2, W1, b1, hb, BS_, FF_, D_, 1);
  gemm_wmma<<<gp, 256, 0, stream>>>(hb, W2, b2, k, BS_, D_, FF_, 0);
  seqnorm<<<gn, 256, 0, stream>>>(k, x2, out);
}